// GNN_node_Virtualnode_16578573763067
// MI455X (gfx1250) — compile-verified
//
#include <hip/hip_runtime.h>
#include <hip/hip_bf16.h>

typedef float v2f __attribute__((ext_vector_type(2)));
typedef float v8f __attribute__((ext_vector_type(8)));

#define D_ 128
#define BN_INV_F 0.9999950000374998f  // 1/sqrt(1+1e-5)

// Guaranteed-hardware fp32 global atomic add (no return -> STOREcnt, no CAS loop).
// Device scope so the L2 RMW is coherent across all WGPs.
__device__ __forceinline__ void gatomic_add_f32(float* p, float v) {
    asm volatile("global_atomic_add_f32 %0, %1, off scope:SCOPE_DEV"
                 :: "v"(p), "v"(v)
                 : "memory");
}

// ---------------- helper kernels ----------------

__global__ __launch_bounds__(256) void k_zero(float4* p, int n4) {
    int i = blockIdx.x * 256 + threadIdx.x;
    if (i < n4) p[i] = make_float4(0.f, 0.f, 0.f, 0.f);
}

__global__ __launch_bounds__(256) void k_copy(const float4* __restrict__ s, float4* __restrict__ d, int n4) {
    int i = blockIdx.x * 256 + threadIdx.x;
    if (i < n4) d[i] = s[i];
}

__global__ __launch_bounds__(256) void k_vn_init(const float* __restrict__ vn_emb,
                                                 float* __restrict__ vn, int G) {
    int i = blockIdx.x * 256 + threadIdx.x;
    if (i < G * D_) vn[i] = vn_emb[i & (D_ - 1)];
}

// h[n,:] = sum_f atom_emb[f, x[n,f], :]   (one wave32 per node, float4 per lane)
__global__ __launch_bounds__(256) void k_atom(const int* __restrict__ x,
                                              const float* __restrict__ aemb,
                                              float* __restrict__ h, int N) {
    int n = blockIdx.x * 8 + (threadIdx.x >> 5);
    if (n >= N) return;
    int lane = threadIdx.x & 31;
    float4 s = make_float4(0.f, 0.f, 0.f, 0.f);
#pragma unroll
    for (int f = 0; f < 9; ++f) {
        int v = x[n * 9 + f];
        float4 tv = ((const float4*)(aemb + (size_t)(f * 64 + v) * D_))[lane];
        s.x += tv.x; s.y += tv.y; s.z += tv.z; s.w += tv.w;
    }
    ((float4*)(h + (size_t)n * D_))[lane] = s;
}

// h[n,:] += vn[batch[n],:]
__global__ __launch_bounds__(256) void k_addvn(float* __restrict__ h,
                                               const float* __restrict__ vn,
                                               const int* __restrict__ batch, int N) {
    int n = blockIdx.x * 8 + (threadIdx.x >> 5);
    if (n >= N) return;
    int lane = threadIdx.x & 31;
    int g = batch[n];
    float4* hp = (float4*)(h + (size_t)n * D_);
    float4 a = hp[lane];
    float4 b = ((const float4*)(vn + (size_t)g * D_))[lane];
    a.x += b.x; a.y += b.y; a.z += b.z; a.w += b.w;
    hp[lane] = a;
}

// per edge: msg = relu(h[src] + sum_f bond_emb[f, ea[f]]); agg[dst] += msg (HW fp32 atomics)
__global__ __launch_bounds__(256) void k_edge(const int* __restrict__ ei,
                                              const int* __restrict__ ea,
                                              const float* __restrict__ bemb,  // [3,8,D] for this layer
                                              const float* __restrict__ h,
                                              float* __restrict__ agg, int E) {
    int e = blockIdx.x * 8 + (threadIdx.x >> 5);
    if (e >= E) return;
    int lane = threadIdx.x & 31;
    int src = ei[e];
    int dst = ei[E + e];
    int a0 = ea[e * 3 + 0], a1 = ea[e * 3 + 1], a2 = ea[e * 3 + 2];
    float4 hv = ((const float4*)(h + (size_t)src * D_))[lane];
    float4 e0 = ((const float4*)(bemb + (size_t)(a0) * D_))[lane];
    float4 e1 = ((const float4*)(bemb + (size_t)(8 + a1) * D_))[lane];
    float4 e2 = ((const float4*)(bemb + (size_t)(16 + a2) * D_))[lane];
    float m0 = fmaxf(hv.x + e0.x + e1.x + e2.x, 0.f);
    float m1 = fmaxf(hv.y + e0.y + e1.y + e2.y, 0.f);
    float m2 = fmaxf(hv.z + e0.z + e1.z + e2.z, 0.f);
    float m3 = fmaxf(hv.w + e0.w + e1.w + e2.w, 0.f);
    float* p = agg + (size_t)dst * D_ + lane * 4;
    gatomic_add_f32(p + 0, m0);
    gatomic_add_f32(p + 1, m1);
    gatomic_add_f32(p + 2, m2);
    gatomic_add_f32(p + 3, m3);
}

// vt[batch[n],:] += h[n,:]
__global__ __launch_bounds__(256) void k_scatter_graph(const float* __restrict__ h,
                                                       const int* __restrict__ batch,
                                                       float* __restrict__ vt, int N) {
    int n = blockIdx.x * 8 + (threadIdx.x >> 5);
    if (n >= N) return;
    int lane = threadIdx.x & 31;
    int g = batch[n];
    float4 hv = ((const float4*)(h + (size_t)n * D_))[lane];
    float* p = vt + (size_t)g * D_ + lane * 4;
    gatomic_add_f32(p + 0, hv.x);
    gatomic_add_f32(p + 1, hv.y);
    gatomic_add_f32(p + 2, hv.z);
    gatomic_add_f32(p + 3, hv.w);
}

// ---------------- fused GEMM + bias + BN (+ReLU) via V_WMMA_F32_16X16X4_F32 ----------------
// out[M,128] = act( ((Aeff @ W) + bias) * (BN_INV*gamma) + beta )
// Aeff = (A2 ? (1+eps)*A + A2 : A).  8 waves/block; wave w -> rows [w*16, w*16+16) x all 128 cols.
__global__ __launch_bounds__(256) void k_gemm(const float* __restrict__ A,
                                              const float* __restrict__ A2,
                                              const float* __restrict__ epsp,
                                              const float* __restrict__ W,
                                              const float* __restrict__ bias,
                                              const float* __restrict__ gamma,
                                              const float* __restrict__ beta,
                                              float* __restrict__ out,
                                              float* __restrict__ out2, int out2_stride,
                                              int M, int relu_flag) {
    __shared__ float sA[128 * 33];                 // 128 rows x 32 K (pitch 33)
    __shared__ __align__(16) float sW[32 * D_];    // 32 K rows x 128 cols

    const int tid = threadIdx.x;
    const int wave = tid >> 5;
    const int lane = tid & 31;
    const int rowBase = blockIdx.x * 128;

    float ep = 1.0f;
    if (A2) ep = 1.0f + epsp[0];

    v8f acc[8];
#pragma unroll
    for (int n = 0; n < 8; ++n) acc[n] = (v8f){0.f, 0.f, 0.f, 0.f, 0.f, 0.f, 0.f, 0.f};

    const int arow = (wave << 4) + (lane & 15);
    const int kh = (lane >> 4) << 1;   // 0 for lanes 0-15, 2 for lanes 16-31

    for (int kc = 0; kc < 4; ++kc) {
        // stage A chunk (rows rowBase..+127, cols kc*32..+31)
#pragma unroll
        for (int i = 0; i < 4; ++i) {
            int idx4 = tid + i * 256;            // 1024 float4 per chunk
            int row = idx4 >> 3;
            int c = (idx4 & 7) << 2;
            int grow = rowBase + row;
            float4 v = make_float4(0.f, 0.f, 0.f, 0.f);
            if (grow < M) {
                v = *(const float4*)(A + (size_t)grow * D_ + kc * 32 + c);
                if (A2) {
                    float4 b = *(const float4*)(A2 + (size_t)grow * D_ + kc * 32 + c);
                    v.x = fmaf(ep, v.x, b.x); v.y = fmaf(ep, v.y, b.y);
                    v.z = fmaf(ep, v.z, b.z); v.w = fmaf(ep, v.w, b.w);
                }
            }
            float* s = sA + row * 33 + c;
            s[0] = v.x; s[1] = v.y; s[2] = v.z; s[3] = v.w;
        }
        // stage W chunk (K rows kc*32..+31, all 128 cols)
#pragma unroll
        for (int i = 0; i < 4; ++i) {
            int idx4 = tid + i * 256;
            int row = idx4 >> 5;
            int c = (idx4 & 31) << 2;
            *(float4*)(sW + row * D_ + c) =
                *(const float4*)(W + (size_t)(kc * 32 + row) * D_ + c);
        }
        __syncthreads();

#pragma unroll
        for (int kk = 0; kk < 8; ++kk) {
            int k = (kk << 2) + kh;
            v2f a;
            a.x = sA[arow * 33 + k];
            a.y = sA[arow * 33 + k + 1];
#pragma unroll
            for (int n = 0; n < 8; ++n) {
                int bcol = (n << 4) + (lane & 15);
                v2f b;
                b.x = sW[k * D_ + bcol];
                b.y = sW[(k + 1) * D_ + bcol];
                acc[n] = __builtin_amdgcn_wmma_f32_16x16x4_f32(
                    false, a, false, b, (short)0, acc[n], false, false);
            }
        }
        __syncthreads();
    }

    // epilogue: bias + BN + optional ReLU; C/D layout: VGPR r -> M=r (lanes0-15), M=r+8 (lanes16-31)
    const int halfsel = lane >> 4;
    const int colIn = lane & 15;
#pragma unroll
    for (int n = 0; n < 8; ++n) {
        int col = (n << 4) + colIn;
        float sc = BN_INV_F * gamma[col];
        float bb = beta[col];
        float bi = bias[col];
#pragma unroll
        for (int r = 0; r < 8; ++r) {
            int row = rowBase + (wave << 4) + r + (halfsel << 3);
            if (row < M) {
                float v = fmaf(acc[n][r] + bi, sc, bb);
                if (relu_flag) v = fmaxf(v, 0.f);
                out[(size_t)row * D_ + col] = v;
                if (out2) out2[(size_t)row * out2_stride + col] = v;
            }
        }
    }
}

// ---------------- orchestration ----------------

extern "C" void kernel_launch(void* const* d_in, const int* in_sizes, int n_in,
                              void* d_out, int out_size, void* d_ws, size_t ws_size,
                              hipStream_t stream) {
    const int*   x          = (const int*)  d_in[0];
    const int*   edge_index = (const int*)  d_in[1];
    const int*   edge_attr  = (const int*)  d_in[2];
    const int*   batch      = (const int*)  d_in[3];
    const float* atom_emb   = (const float*)d_in[4];
    const float* vn_emb     = (const float*)d_in[5];
    const float* bond_emb   = (const float*)d_in[6];
    const float* eps        = (const float*)d_in[7];
    const float* conv_w1    = (const float*)d_in[8];
    const float* conv_b1    = (const float*)d_in[9];
    const float* conv_bn_g  = (const float*)d_in[10];
    const float* conv_bn_b  = (const float*)d_in[11];
    const float* conv_w2    = (const float*)d_in[12];
    const float* conv_b2    = (const float*)d_in[13];
    const float* bn_g       = (const float*)d_in[14];
    const float* bn_b       = (const float*)d_in[15];
    const float* vn_w1      = (const float*)d_in[16];
    const float* vn_b1      = (const float*)d_in[17];
    const float* vn_bn1_g   = (const float*)d_in[18];
    const float* vn_bn1_b   = (const float*)d_in[19];
    const float* vn_w2      = (const float*)d_in[20];
    const float* vn_b2      = (const float*)d_in[21];
    const float* vn_bn2_g   = (const float*)d_in[22];
    const float* vn_bn2_b   = (const float*)d_in[23];

    const int N = in_sizes[3];
    const int E = in_sizes[1] / 2;
    const int L = in_sizes[7];
    const int G = (out_size - N * D_) / ((L - 1) * D_);

    float* ws   = (float*)d_ws;
    float* hA   = ws;  ws += (size_t)N * D_;
    float* hB   = ws;  ws += (size_t)N * D_;
    float* agg  = ws;  ws += (size_t)N * D_;   // also reused as t1 (safe: per-block rows read-before-write)
    float* vn   = ws;  ws += (size_t)G * D_;
    float* vt   = ws;  ws += (size_t)G * D_;
    float* vtmp = ws;  ws += (size_t)G * D_;
    float* t1 = agg;
    float* out = (float*)d_out;
    float* fp  = out + (size_t)N * D_;

    dim3 blk(256);
    int nodeBlocks = (N + 7) / 8;
    int edgeBlocks = (E + 7) / 8;
    int nd4 = N * D_ / 4;
    int gd4 = G * D_ / 4;
    int gemmBlocksN = (N + 127) / 128;
    int gemmBlocksG = (G + 127) / 128;

    k_atom<<<nodeBlocks, blk, 0, stream>>>(x, atom_emb, hA, N);
    k_vn_init<<<(G * D_ + 255) / 256, blk, 0, stream>>>(vn_emb, vn, G);

    for (int l = 0; l < L; ++l) {
        k_addvn<<<nodeBlocks, blk, 0, stream>>>(hA, vn, batch, N);
        k_zero<<<(nd4 + 255) / 256, blk, 0, stream>>>((float4*)agg, nd4);
        k_edge<<<edgeBlocks, blk, 0, stream>>>(edge_index, edge_attr,
                                               bond_emb + (size_t)l * 3 * 8 * D_,
                                               hA, agg, E);
        // t1 = relu(bn(((1+eps)h + agg) @ w1 + b1))
        k_gemm<<<gemmBlocksN, blk, 0, stream>>>(hA, agg, eps + l,
                                                conv_w1 + (size_t)l * D_ * D_, conv_b1 + l * D_,
                                                conv_bn_g + l * D_, conv_bn_b + l * D_,
                                                t1, nullptr, 0, N, 1);
        // hB/out = bn(t1 @ w2 + b2) (+relu except last layer)
        float* o2 = (l == L - 1) ? out : hB;
        k_gemm<<<gemmBlocksN, blk, 0, stream>>>(t1, nullptr, nullptr,
                                                conv_w2 + (size_t)l * D_ * D_, conv_b2 + l * D_,
                                                bn_g + l * D_, bn_b + l * D_,
                                                o2, nullptr, 0, N, (l < L - 1) ? 1 : 0);
        if (l < L - 1) {
            // vt = vn + segment_sum(h, batch)   (pre-conv h)
            k_copy<<<(gd4 + 255) / 256, blk, 0, stream>>>((const float4*)vn, (float4*)vt, gd4);
            k_scatter_graph<<<nodeBlocks, blk, 0, stream>>>(hA, batch, vt, N);
            k_gemm<<<gemmBlocksG, blk, 0, stream>>>(vt, nullptr, nullptr,
                                                    vn_w1 + (size_t)l * D_ * D_, vn_b1 + l * D_,
                                                    vn_bn1_g + l * D_, vn_bn1_b + l * D_,
                                                    vtmp, nullptr, 0, G, 1);
            k_gemm<<<gemmBlocksG, blk, 0, stream>>>(vtmp, nullptr, nullptr,
                                                    vn_w2 + (size_t)l * D_ * D_, vn_b2 + l * D_,
                                                    vn_bn2_g + l * D_, vn_bn2_b + l * D_,
                                                    vn, fp + (size_t)l * D_, (L - 1) * D_, G, 1);
            float* tmp = hA; hA = hB; hB = tmp;
        }
    }
}